// BranchDiversity_6760278524657
// MI455X (gfx1250) — compile-verified
//
#include <hip/hip_runtime.h>
#include <math.h>

typedef __attribute__((ext_vector_type(2))) float v2f;
typedef __attribute__((ext_vector_type(8))) float v8f;

#define NBR   8      // branches
#define BATCH 128
#define CHN   512
#define HW    49     // 7*7
#define KPAD  52     // 49 padded to multiple of 4 for 16x16x4 WMMA k-loop
#define GAMMA 10.0f

// ---------------------------------------------------------------------------
// Kernel 1: channel mean.  x[nb][c][hw] -> xm[nb][hw] = mean over c.
// One block per (branch,batch) slab (512*49 contiguous floats).
// Lane dimension = hw (coalesced), 8 channel-slices of 64 for MLP,
// cross-slice reduction via LDS.  Pure-bandwidth kernel: reads 103 MB once.
// ---------------------------------------------------------------------------
__global__ __launch_bounds__(512) void chan_mean_kernel(const float* __restrict__ x,
                                                        float* __restrict__ xm) {
    __shared__ float part[8][HW];
    const int nb = blockIdx.x;          // 0..1023
    const int t  = threadIdx.x;         // 0..511
    const int hw = t & 63;              // lane-contiguous -> coalesced
    const int cg = t >> 6;              // channel slice 0..7 (64 channels each)
    const float* base = x + (size_t)nb * (CHN * HW);
    if (hw < HW) {
        const float* p = base + (size_t)cg * 64 * HW + hw;
        float s = 0.0f;
        #pragma unroll 8
        for (int c = 0; c < 64; ++c) s += p[c * HW];
        part[cg][hw] = s;
    }
    __syncthreads();
    if (t < HW) {
        float acc = 0.0f;
        #pragma unroll
        for (int g = 0; g < 8; ++g) acc += part[g][t];
        xm[(size_t)nb * HW + t] = acc * (1.0f / CHN);
    }
}

// ---------------------------------------------------------------------------
// Kernel 2: per-batch Gram matrix via V_WMMA_F32_16X16X4_F32.
// One wave per batch b.  M[i][p] = xm[i,b,p] (8x49, zero-padded to 16x52).
// G = M * M^T accumulated over 13 k-steps of 4.  sq = n_i + n_j - 2 G[i][j];
// partial[b][i][j] = (i==j) ? 0 : exp(-GAMMA*sq) / BATCH.
// A and B fragments are identical per lane (B = M^T, same LDS reads).
// ---------------------------------------------------------------------------
__global__ __launch_bounds__(32) void pairwise_wmma_kernel(const float* __restrict__ xm,
                                                           float* __restrict__ part) {
    __shared__ float M[16][KPAD];
    __shared__ float nrm[NBR];
    const int b = blockIdx.x;           // 0..127
    const int l = threadIdx.x;          // 0..31

    // Fill padded tile (rows >=8 and cols >=49 are zero).
    for (int idx = l; idx < 16 * KPAD; idx += 32) {
        const int i = idx / KPAD;
        const int p = idx % KPAD;
        float v = 0.0f;
        if (i < NBR && p < HW) v = xm[((size_t)i * BATCH + b) * HW + p];
        M[i][p] = v;
    }
    __syncthreads();

    if (l < NBR) {
        float s = 0.0f;
        for (int p = 0; p < HW; ++p) { const float v = M[l][p]; s += v * v; }
        nrm[l] = s;
    }
    __syncthreads();

    // Build fragments per ISA layout:
    //   A 16x4 f32: lane l -> row m = l%16; vgpr0/1 hold k0/k0+1, k0 = 4kb + (l>=16 ? 2 : 0)
    //   B 4x16 f32: lane l -> col n = l%16; same k striping -> identical data for B = M^T
    const int m     = l & 15;
    const int hiOff = (l >= 16) ? 2 : 0;
    v8f acc = {};
    #pragma unroll
    for (int kb = 0; kb < KPAD / 4; ++kb) {
        const int k0 = kb * 4 + hiOff;
        v2f a;
        a.x = M[m][k0];
        a.y = M[m][k0 + 1];
        acc = __builtin_amdgcn_wmma_f32_16x16x4_f32(
            /*neg_a=*/false, a, /*neg_b=*/false, a,
            /*c_mod=*/(short)0, acc, /*reuse_a=*/false, /*reuse_b=*/false);
    }

    // C/D layout: vgpr r, lane l<16 -> G[r][l].  Lanes 0..7 hold column j=l,
    // rows 0..7 in acc[0..7] — exactly the useful 8x8 block.
    if (l < NBR) {
        const int j = l;
        const float nj = nrm[j];
        float* dst = part + (size_t)b * (NBR * NBR);
        #pragma unroll
        for (int i = 0; i < NBR; ++i) {
            const float sq = nrm[i] + nj - 2.0f * acc[i];
            const float e  = (i == j) ? 0.0f : expf(-GAMMA * sq);
            dst[i * NBR + j] = e * (1.0f / BATCH);
        }
    }
}

// ---------------------------------------------------------------------------
// Kernel 3: deterministic reduction over batches + det / slogdet of 8x8.
// ---------------------------------------------------------------------------
__global__ __launch_bounds__(64) void finalize_kernel(const float* __restrict__ part,
                                                      float* __restrict__ out) {
    __shared__ float snm[NBR * NBR];
    const int t = threadIdx.x;          // 0..63, one (i,j) entry each
    float s = 0.0f;
    for (int b = 0; b < BATCH; ++b) s += part[(size_t)b * 64 + t];  // fixed order
    snm[t] = s;
    __syncthreads();

    if (t == 0) {
        // direct_div = sum of all entries (diagonal already zero)
        float direct = 0.0f;
        for (int k = 0; k < 64; ++k) direct += snm[k];
        out[0] = direct;

        // LU with partial pivoting in double for det and slogdet
        double A[8][8];
        for (int i = 0; i < 8; ++i)
            for (int j = 0; j < 8; ++j) A[i][j] = (double)snm[i * 8 + j];

        double det = 1.0;
        int sign = 1;
        for (int c = 0; c < 8; ++c) {
            int piv = c;
            double mx = fabs(A[c][c]);
            for (int r = c + 1; r < 8; ++r) {
                const double v = fabs(A[r][c]);
                if (v > mx) { mx = v; piv = r; }
            }
            if (mx == 0.0) { det = 0.0; break; }
            if (piv != c) {
                for (int j = 0; j < 8; ++j) {
                    const double tmp = A[c][j]; A[c][j] = A[piv][j]; A[piv][j] = tmp;
                }
                sign = -sign;
            }
            det *= A[c][c];
            const double inv = 1.0 / A[c][c];
            for (int r = c + 1; r < 8; ++r) {
                const double f = A[r][c] * inv;
                for (int j = c; j < 8; ++j) A[r][j] -= f * A[c][j];
            }
        }
        det *= (double)sign;

        out[1] = (float)(-det);                               // det_div
        out[2] = (det > 0.0) ? (float)(-log(det))             // logdet_div
                             : __int_as_float(0x7fc00000);    // NaN (sign <= 0)
    }
}

// ---------------------------------------------------------------------------
extern "C" void kernel_launch(void* const* d_in, const int* in_sizes, int n_in,
                              void* d_out, int out_size, void* d_ws, size_t ws_size,
                              hipStream_t stream) {
    const float* x = (const float*)d_in[0];
    float* out = (float*)d_out;

    float* xm   = (float*)d_ws;                       // 1024*49   = 50176 floats
    float* part = xm + (size_t)NBR * BATCH * HW;      // 128*64    =  8192 floats

    chan_mean_kernel<<<NBR * BATCH, 512, 0, stream>>>(x, xm);
    pairwise_wmma_kernel<<<BATCH, 32, 0, stream>>>(xm, part);
    finalize_kernel<<<1, 64, 0, stream>>>(part, out);
}